// LightGCN_35802847380042
// MI455X (gfx1250) — compile-verified
//
#include <hip/hip_runtime.h>
#include <hip/hip_bf16.h>

#define NUM_USERS 200000
#define NUM_ITEMS 400000
#define N_NODES   (NUM_USERS + NUM_ITEMS)   // 600000
#define EMB_DIM   64
#define NUM_LAYERS 3
#define N_EDGES   1200000

typedef __attribute__((ext_vector_type(2))) float v2f;
typedef __attribute__((ext_vector_type(8))) float v8f;

// ---------------- degree histogram: deg[col]++ ----------------
__global__ void k_degree(const long long* __restrict__ eidx, unsigned* __restrict__ deg) {
    long t = (long)blockIdx.x * blockDim.x + threadIdx.x;
    if (t >= N_EDGES) return;
    int col = (int)eidx[N_EDGES + t];
    atomicAdd(&deg[col], 1u);           // global_atomic_add_u32
}

// ---------------- deg_inv_sqrt ----------------
__global__ void k_dis(const unsigned* __restrict__ deg, float* __restrict__ dis) {
    long t = (long)blockIdx.x * blockDim.x + threadIdx.x;
    if (t >= N_NODES) return;
    unsigned d = deg[t];
    dis[t] = (d > 0u) ? (1.0f / sqrtf((float)d)) : 0.0f;
}

// ------- pack int64 indices -> int32 and precompute edge norm -------
__global__ void k_prep(const long long* __restrict__ eidx, const float* __restrict__ dis,
                       int* __restrict__ row32, int* __restrict__ col32,
                       float* __restrict__ norm) {
    long t = (long)blockIdx.x * blockDim.x + threadIdx.x;
    if (t >= N_EDGES) return;
    int r = (int)eidx[t];
    int c = (int)eidx[N_EDGES + t];
    row32[t] = r;
    col32[t] = c;
    norm[t]  = dis[r] * dis[c];
}

// ------- one propagation layer: x_new[col] += norm * x[row] -------
// 16 lanes per edge, one float4 (16 B) per lane -> full 256 B row coverage.
__global__ void k_scatter(const float* __restrict__ x,
                          const int* __restrict__ row32, const int* __restrict__ col32,
                          const float* __restrict__ norm, float* __restrict__ xnew) {
    long tid = (long)blockIdx.x * blockDim.x + threadIdx.x;
    long e = tid >> 4;
    if (e >= N_EDGES) return;
    int c4 = (int)(tid & 15) * 4;
    int r  = row32[e];
    int cl = col32[e];
    float w = norm[e];
    const float4 v = *(const float4*)(x + (long)r * EMB_DIM + c4);
    float* dst = xnew + (long)cl * EMB_DIM + c4;
    unsafeAtomicAdd(dst + 0, w * v.x);  // global_atomic_add_f32
    unsafeAtomicAdd(dst + 1, w * v.y);
    unsafeAtomicAdd(dst + 2, w * v.z);
    unsafeAtomicAdd(dst + 3, w * v.w);
}

// ---------------- acc += x_new (float4) ----------------
__global__ void k_acc(float* __restrict__ out, const float* __restrict__ xn) {
    long t = (long)blockIdx.x * blockDim.x + threadIdx.x;   // N_NODES*EMB_DIM/4 threads
    float4 a = ((const float4*)out)[t];
    float4 b = ((const float4*)xn)[t];
    a.x += b.x; a.y += b.y; a.z += b.z; a.w += b.w;
    ((float4*)out)[t] = a;
}

// ---------------- finalize: out = 0.25 * out, via V_WMMA_F32_16X16X4_F32 ----
// One wave per 16x16 tile. D = sum_c A_c(16x4) x B_c(4x16), where A_c are the
// 4-column chunks of the tile and B_c[k][n] = 0.25 * delta(4c+k, n). Exactly
// one nonzero product per output element and 0.25 is a power of two, so the
// result is bit-exact 0.25*tile in fp32.
// Layouts (ISA 7.12.2): A 16x4 f32: lanes 0-15 hold M=lane, VGPR j -> K = 2*(lane/16)+j.
// B mirrored with N=lane%16. C/D: lane L, VGPR v -> (M = v + 8*(L/16), N = L%16).
__global__ void k_finalize(float* __restrict__ out) {
    const int lane = threadIdx.x & 31;
    const int wave = threadIdx.x >> 5;
    const int tile = blockIdx.x * (blockDim.x >> 5) + wave;   // exact grid: no guard
    const int tilesPerRow = EMB_DIM / 16;                     // 4
    const long nt = tile / tilesPerRow;                       // node-tile
    const int  dt = tile % tilesPerRow;                       // dim-tile
    const int  m  = lane & 15;
    const int  hi = lane >> 4;                                // 0 or 1
    const int  k0 = 2 * hi;

    const float* rowp = out + (nt * 16 + m) * EMB_DIM + dt * 16;

    v8f d = {};
#pragma unroll
    for (int c = 0; c < 4; ++c) {
        v2f a, b;
        a.x = rowp[4 * c + k0 + 0];                 // A_c[m][k0]
        a.y = rowp[4 * c + k0 + 1];                 // A_c[m][k0+1]
        b.x = (m == 4 * c + k0 + 0) ? 0.25f : 0.0f; // B_c[k0][m]
        b.y = (m == 4 * c + k0 + 1) ? 0.25f : 0.0f; // B_c[k0+1][m]
        d = __builtin_amdgcn_wmma_f32_16x16x4_f32(
                /*neg_a=*/false, a, /*neg_b=*/false, b,
                /*c_mod=*/(short)0, d, /*reuse_a=*/false, /*reuse_b=*/false);
    }
#pragma unroll
    for (int v = 0; v < 8; ++v) {
        out[(nt * 16 + v + 8 * hi) * EMB_DIM + dt * 16 + m] = d[v];
    }
}

extern "C" void kernel_launch(void* const* d_in, const int* in_sizes, int n_in,
                              void* d_out, int out_size, void* d_ws, size_t ws_size,
                              hipStream_t stream) {
    const float*     emb  = (const float*)d_in[0];
    const long long* eidx = (const long long*)d_in[1];      // int64 [2, N_EDGES]
    float* out = (float*)d_out;

    // workspace carve-up (all region sizes are multiples of 256 B)
    char* ws = (char*)d_ws;
    size_t off = 0;
    unsigned* deg   = (unsigned*)(ws + off); off += (size_t)N_NODES * 4;
    float*    dis   = (float*)   (ws + off); off += (size_t)N_NODES * 4;
    float*    norm  = (float*)   (ws + off); off += (size_t)N_EDGES * 4;
    int*      row32 = (int*)     (ws + off); off += (size_t)N_EDGES * 4;
    int*      col32 = (int*)     (ws + off); off += (size_t)N_EDGES * 4;
    const size_t xbytes = (size_t)N_NODES * EMB_DIM * 4;    // 153.6 MB
    float*    xa    = (float*)   (ws + off); off += xbytes;
    float*    xb    = (float*)   (ws + off); off += xbytes;

    // 1) degree + normalization coefficients
    hipMemsetAsync(deg, 0, (size_t)N_NODES * 4, stream);
    k_degree<<<(N_EDGES + 255) / 256, 256, 0, stream>>>(eidx, deg);
    k_dis   <<<(N_NODES + 255) / 256, 256, 0, stream>>>(deg, dis);
    k_prep  <<<(N_EDGES + 255) / 256, 256, 0, stream>>>(eidx, dis, row32, col32, norm);

    // 2) acc starts as the embedding (accumulated directly in d_out)
    hipMemcpyAsync(out, emb, xbytes, hipMemcpyDeviceToDevice, stream);

    // 3) three propagation layers, ping-ponging xa/xb
    const long scat_threads = (long)N_EDGES * 16;           // 19.2 M
    const int  scat_blocks  = (int)(scat_threads / 256);    // 75000 exact
    const int  acc_blocks   = (int)((size_t)N_NODES * EMB_DIM / 4 / 256); // 37500 exact

    const float* xcur = emb;
    float* bufs[2] = { xa, xb };
    for (int l = 0; l < NUM_LAYERS; ++l) {
        float* xn = bufs[l & 1];
        hipMemsetAsync(xn, 0, xbytes, stream);
        k_scatter<<<scat_blocks, 256, 0, stream>>>(xcur, row32, col32, norm, xn);
        k_acc    <<<acc_blocks, 256, 0, stream>>>(out, xn);
        xcur = xn;
    }

    // 4) out *= 0.25 on the matrix pipe (16x16 tiles, one wave each)
    const int n_tiles     = (N_NODES / 16) * (EMB_DIM / 16); // 150000
    const int fin_blocks  = n_tiles / 8;                     // 8 waves/block, 18750 exact
    k_finalize<<<fin_blocks, 256, 0, stream>>>(out);
}